// VQVAE_42013370090102
// MI455X (gfx1250) — compile-verified
//
#include <hip/hip_runtime.h>
#include <hip/hip_bf16.h>

typedef __attribute__((ext_vector_type(2))) float v2f;
typedef __attribute__((ext_vector_type(8))) float v8f;

__device__ __forceinline__ int iclamp(int x, int lo, int hi) {
  return x < lo ? lo : (x > hi ? hi : x);
}

// ---------------------------------------------------------------------------
// conv1d / conv_transpose1d as implicit GEMM on fp32 WMMA (v_wmma_f32_16x16x4).
// Block = 4 waves sharing one 32-col N-group over 256 M-rows; per-wave tile is
// 64(M) x 32(N) (8 accumulators). Weights (R x 32, R = Cin*Kw <= 512, <=64KB)
// are cooperatively staged into LDS once per block in (r-pair, col) order so
// each lane's WMMA K-pair is one aligned ds_load_b64. Per K-step: 8 global A
// loads + 2 ds_load_b64 feed 8 WMMAs. Reduction order r = k*Cin + ci; spatial
// validity hoisted to once-per-k; ci loop is running pointers + selects.
// ---------------------------------------------------------------------------
__global__ __launch_bounds__(128) void conv1d_wmma_kernel(
    const float* __restrict__ in, const float* __restrict__ w,
    const float* __restrict__ bias, const float* __restrict__ resid,
    float* __restrict__ out,
    int B, int Cin, int Lin, int Cout, int Lout,
    int Kw, int stride, int pad, int inBS, int outBS,
    int relu, int transposed)
{
  extern __shared__ float sB[];                 // (R/2) * 64 floats
  const int lane = threadIdx.x;                 // 0..31 (wave32)
  const int wv   = threadIdx.y;                 // 0..3  wave id in block
  const int ntG  = (Cout + 31) >> 5;
  const int mb   = blockIdx.x / ntG;
  const int ng   = blockIdx.x - mb * ntG;       // shared N-group (32 cols)
  const int Mtot = B * Lout;
  const int mt   = mb * 4 + wv;                 // this wave's 64-row tile

  const int half = lane >> 4;                   // 0/1: selects K pair
  const int lid  = lane & 15;
  const int R    = Cin * Kw;

  int col[2];
#pragma unroll
  for (int n = 0; n < 2; ++n) col[n] = ng * 32 + n * 16 + lid;

  int at[4];
  const float* inP[4];
#pragma unroll
  for (int m = 0; m < 4; ++m) {
    int rowm = mt * 64 + m * 16 + lid;          // A row = b*Lout + t
    int rowc = rowm < Mtot ? rowm : Mtot - 1;   // clamp: padding rows compute
    int ab   = rowc / Lout;                     //        garbage, never stored
    at[m]    = rowc - ab * Lout;
    inP[m]   = in + (size_t)ab * inBS;
  }

  v8f acc[4][2];
#pragma unroll
  for (int m = 0; m < 4; ++m)
#pragma unroll
    for (int n = 0; n < 2; ++n)
      acc[m][n] = (v8f){0.f, 0.f, 0.f, 0.f, 0.f, 0.f, 0.f, 0.f};

  if ((Cin & 3) == 0) {
    // ---- cooperative weight staging: pairs (2*rp, 2*rp+1) x 32 cols ----
    {
      const int tid = wv * 32 + lane;
      const int npairs = (R >> 1) * 32;
      for (int i = tid; i < npairs; i += 128) {
        int rp = i >> 5, c = i & 31;
        int colG = iclamp(ng * 32 + c, 0, Cout - 1);
        int r0 = rp << 1, r1 = r0 + 1;
        int k0 = r0 / Cin, ci0 = r0 - k0 * Cin;   // r = k*Cin + ci
        int k1 = r1 / Cin, ci1 = r1 - k1 * Cin;
        float w0, w1;
        if (!transposed) {                        // w[Cout][Cin][Kw]
          w0 = w[((size_t)colG * Cin + ci0) * Kw + k0];
          w1 = w[((size_t)colG * Cin + ci1) * Kw + k1];
        } else {                                  // w[Cin][Cout][Kw]
          w0 = w[((size_t)ci0 * Cout + colG) * Kw + k0];
          w1 = w[((size_t)ci1 * Cout + colG) * Kw + k1];
        }
        sB[rp * 64 + c * 2]     = w0;
        sB[rp * 64 + c * 2 + 1] = w1;
      }
    }
    __syncthreads();

    // ---- main loop: Cin multiple of 4, reduction r = k*Cin + ci ----
    for (int k = 0; k < Kw; ++k) {
      const float* pa[4];
      bool va[4];
#pragma unroll
      for (int m = 0; m < 4; ++m) {
        int p;
        if (!transposed) {
          p = at[m] * stride + k - pad;
          va[m] = (p >= 0) && (p < Lin);
        } else {
          int mm = at[m] + pad - k;
          int jj = mm >= 0 ? mm / stride : 0;   // once per k: cheap
          va[m] = (mm >= 0) && (mm - jj * stride == 0) && (jj < Lin);
          p = jj;
        }
        pa[m] = inP[m] + iclamp(p, 0, Lin - 1) + (size_t)(half * 2) * Lin;
      }
      unsigned off0 = (unsigned)(((k * Cin) >> 1) + half) * 64 + lid * 2;
      unsigned off1 = off0 + 32;                // n=1: col+16 -> +32 floats

      for (int ci0 = 0; ci0 < Cin; ci0 += 4) {
        v2f bv0 = *(const v2f*)(sB + off0);     // ds_load_b64
        v2f bv1 = *(const v2f*)(sB + off1);     // ds_load_b64
        off0 += 128; off1 += 128;
#pragma unroll
        for (int m = 0; m < 4; ++m) {
          float a0 = pa[m][0];
          float a1 = pa[m][Lin];
          pa[m] += (size_t)4 * Lin;
          v2f av = { va[m] ? a0 : 0.0f, va[m] ? a1 : 0.0f };
          acc[m][0] = __builtin_amdgcn_wmma_f32_16x16x4_f32(
              false, av, false, bv0, (short)0, acc[m][0], false, false);
          acc[m][1] = __builtin_amdgcn_wmma_f32_16x16x4_f32(
              false, av, false, bv1, (short)0, acc[m][1], false, false);
        }
      }
    }
  } else {
    // ---- Cin==1 path (first layer only): reduction index r == k ----
    for (int k0 = 0; k0 < Kw; k0 += 4) {
      int k = k0 + half * 2;
      bool bk0 = k < Kw, bk1 = (k + 1) < Kw;
      v2f bv[2];
#pragma unroll
      for (int n = 0; n < 2; ++n) {
        int colG = iclamp(col[n], 0, Cout - 1);
        float w0 = w[(size_t)colG * Kw + iclamp(k, 0, Kw - 1)];
        float w1 = w[(size_t)colG * Kw + iclamp(k + 1, 0, Kw - 1)];
        bv[n] = (v2f){ bk0 ? w0 : 0.0f, bk1 ? w1 : 0.0f };
      }
#pragma unroll
      for (int m = 0; m < 4; ++m) {
        int p0 = at[m] * stride + k - pad;
        int p1 = p0 + 1;
        bool v0 = bk0 && (p0 >= 0) && (p0 < Lin);
        bool v1 = bk1 && (p1 >= 0) && (p1 < Lin);
        float a0 = inP[m][iclamp(p0, 0, Lin - 1)];
        float a1 = inP[m][iclamp(p1, 0, Lin - 1)];
        v2f av = { v0 ? a0 : 0.0f, v1 ? a1 : 0.0f };
        acc[m][0] = __builtin_amdgcn_wmma_f32_16x16x4_f32(
            false, av, false, bv[0], (short)0, acc[m][0], false, false);
        acc[m][1] = __builtin_amdgcn_wmma_f32_16x16x4_f32(
            false, av, false, bv[1], (short)0, acc[m][1], false, false);
      }
    }
  }

  // ---- store: C/D layout VGPR v, lane l -> M = v + 8*(l/16), N = l%16 ----
#pragma unroll
  for (int n = 0; n < 2; ++n) {
    const bool cok  = col[n] < Cout;
    const float bval = cok ? bias[col[n]] : 0.0f;
#pragma unroll
    for (int m = 0; m < 4; ++m) {
#pragma unroll
      for (int v = 0; v < 8; ++v) {
        int orow = mt * 64 + m * 16 + v + half * 8;
        if (orow < Mtot && cok) {
          int ob = orow / Lout, ot = orow - ob * Lout;
          float val = acc[m][n][v] + bval;
          if (relu) val = fmaxf(val, 0.0f);
          size_t idx = (size_t)ob * outBS + (size_t)col[n] * Lout + ot;
          if (resid) val += resid[idx];
          out[idx] = val;
        }
      }
    }
  }
}

// ---------------------------------------------------------------------------
// Vector quantize: per (b,t) row of 64-dim vector, argmin L2 over 512 codes,
// scatter chosen code back in NCL layout, accumulate mean squared diff.
// ---------------------------------------------------------------------------
__global__ __launch_bounds__(256) void quantize_kernel(
    const float* __restrict__ qin, const float* __restrict__ embed,
    float* __restrict__ qout, float* __restrict__ diff,
    int B, int L, int inBS, int outBS, float invCount)
{
  int row = blockIdx.x * blockDim.x + threadIdx.x;
  if (row >= B * L) return;
  int b = row / L, t = row - b * L;
  const float* base = qin + (size_t)b * inBS + t;
  float v[64];
#pragma unroll
  for (int d = 0; d < 64; ++d) v[d] = base[(size_t)d * L];

  float best = 3.4e38f; int bi = 0;
  for (int e = 0; e < 512; ++e) {
    const float* E = embed + (size_t)e * 64;
    float s = 0.0f;
#pragma unroll
    for (int d = 0; d < 64; ++d) { float df = v[d] - E[d]; s = fmaf(df, df, s); }
    if (s < best) { best = s; bi = e; }
  }
  const float* E = embed + (size_t)bi * 64;
  float* o = qout + (size_t)b * outBS + t;
#pragma unroll
  for (int d = 0; d < 64; ++d) o[(size_t)d * L] = E[d];
  atomicAdd(diff, best * invCount);
}

__global__ void zero_scalar_kernel(float* p) { *p = 0.0f; }

// ---------------------------------------------------------------------------
// Host-side helpers
// ---------------------------------------------------------------------------
static void conv_launch(hipStream_t st, const float* in, const float* w,
                        const float* bias, const float* resid, float* out,
                        int B, int Cin, int Lin, int Cout, int Lout,
                        int Kw, int stride, int pad, int inBS, int outBS,
                        int relu, int transposed)
{
  int Mtot = B * Lout;
  int mBlocks = (Mtot + 255) / 256;             // 256 rows per block (4 waves)
  int ntG = (Cout + 31) / 32;
  int blocks = mBlocks * ntG;
  size_t shbytes = ((Cin & 3) == 0) ? (size_t)Cin * Kw * 128 : 0;  // R*32 floats
  conv1d_wmma_kernel<<<dim3(blocks), dim3(32, 4, 1), shbytes, st>>>(
      in, w, bias, resid, out, B, Cin, Lin, Cout, Lout,
      Kw, stride, pad, inBS, outBS, relu, transposed);
}

// res_block: h = relu(conv k3 p1, 128->32); x += conv k1 p0, 32->128 (in place)
static void res_block(hipStream_t st, float* xbuf, int xBS,
                      const float* w1, const float* b1,
                      const float* w2, const float* b2,
                      float* h, int B, int L)
{
  conv_launch(st, xbuf, w1, b1, nullptr, h, B, 128, L, 32, L, 3, 1, 1, xBS, 32 * L, 1, 0);
  conv_launch(st, h, w2, b2, xbuf, xbuf, B, 32, L, 128, L, 1, 1, 0, 32 * L, xBS, 0, 0);
}

extern "C" void kernel_launch(void* const* d_in, const int* in_sizes, int n_in,
                              void* d_out, int out_size, void* d_ws, size_t ws_size,
                              hipStream_t stream)
{
  (void)in_sizes; (void)n_in; (void)out_size; (void)ws_size;
  const int B = 16, Lx = 16384, Lb = 4096, Lt = 2048, Ld = 16382;

  // Input leaves, setup_inputs() insertion-order flattening:
  // 0:x 1:enc_b.w 2:enc_b.b 3-10:enc_b.res 11:enc_t.w 12:enc_t.b 13-20:enc_t.res
  // 21:qconv_t.w 22:qconv_t.b 23:embed_t 24:dec_t.wT 25:dec_t.bT 26-33:dec_t.res
  // 34:dec_t.wo 35:dec_t.bo 36:qconv_b.w 37:qconv_b.b 38:embed_b
  // 39:upsample_t.wT 40:upsample_t.bT 41:dec.wT 42:dec.bT 43-50:dec.res
  // 51:dec.wo 52:dec.bo
#define F(i) ((const float*)d_in[(i)])
  const float* x = F(0);

  // Workspace layout (floats), lifetimes overlapped; total 44,036,096 floats.
  float* ws      = (float*)d_ws;
  float* cat     = ws + 0;            // (B,192,Lb): ch0-63 dec_t out, ch64-191 enc_b
  float* qb      = ws + 12582912;     // (B, 64,Lb)
  float* e_t     = ws + 16777216;     // (B,128,Lt)
  float* qt      = ws + 20971520;     // (B, 64,Lt)
  float* d1      = ws + 23068672;     // (B,128,Lb)
  float* d2      = ws + 0;            // (B,128,Ld) overlays cat/qb/e_t/qt/d1 (dead)
  float* quant_t = ws + 33550336;     // (B, 64,Lt)
  float* r2      = ws + 35647488;     // (B,128,Lb): res-temp early, quant-concat late
  float* out     = (float*)d_out;     // (B,1,Ld) + diff scalar
  float* diff    = out + (size_t)B * Ld;   // index 262112

  zero_scalar_kernel<<<1, 1, 0, stream>>>(diff);

  // ---- enc_b: conv 1->128 k4 s4 p1, then 2 res blocks (lives inside cat) ----
  float* e_b = cat + 64 * Lb;         // channel slice 64..191 of cat
  const int ebBS = 192 * Lb;
  conv_launch(stream, x, F(1), F(2), nullptr, e_b, B, 1, Lx, 128, Lb, 4, 4, 1, Lx, ebBS, 0, 0);
  res_block(stream, e_b, ebBS, F(3), F(4), F(5), F(6), r2, B, Lb);
  res_block(stream, e_b, ebBS, F(7), F(8), F(9), F(10), r2, B, Lb);

  // ---- enc_t: conv 128->128 k4 s2 p1, 2 res blocks ----
  conv_launch(stream, e_b, F(11), F(12), nullptr, e_t, B, 128, Lb, 128, Lt, 4, 2, 1, ebBS, 128 * Lt, 0, 0);
  res_block(stream, e_t, 128 * Lt, F(13), F(14), F(15), F(16), r2, B, Lt);
  res_block(stream, e_t, 128 * Lt, F(17), F(18), F(19), F(20), r2, B, Lt);

  // ---- qconv_t (128->64 k1) + quantize_t ----
  conv_launch(stream, e_t, F(21), F(22), nullptr, qt, B, 128, Lt, 64, Lt, 1, 1, 0, 128 * Lt, 64 * Lt, 0, 0);
  {
    int rows = B * Lt;
    quantize_kernel<<<(rows + 255) / 256, 256, 0, stream>>>(
        qt, F(23), quant_t, diff, B, Lt, 64 * Lt, 64 * Lt, 1.0f / ((float)rows * 64.0f));
  }

  // ---- dec_t: convT 64->128 k4 s2 p1; 2 res blocks; wo conv 128->64 k3 p1 ----
  conv_launch(stream, quant_t, F(24), F(25), nullptr, d1, B, 64, Lt, 128, Lb, 4, 2, 1, 64 * Lt, 128 * Lb, 0, 1);
  res_block(stream, d1, 128 * Lb, F(26), F(27), F(28), F(29), r2, B, Lb);
  res_block(stream, d1, 128 * Lb, F(30), F(31), F(32), F(33), r2, B, Lb);
  conv_launch(stream, d1, F(34), F(35), nullptr, cat /*ch 0..63*/, B, 128, Lb, 64, Lb, 3, 1, 1, 128 * Lb, 192 * Lb, 0, 0);

  // ---- qconv_b (192->64 k1 over concat) + quantize_b -> quant-concat ch64-127 ----
  conv_launch(stream, cat, F(36), F(37), nullptr, qb, B, 192, Lb, 64, Lb, 1, 1, 0, 192 * Lb, 64 * Lb, 0, 0);
  {
    int rows = B * Lb;
    quantize_kernel<<<(rows + 255) / 256, 256, 0, stream>>>(
        qb, F(38), r2 + 64 * Lb, diff, B, Lb, 64 * Lb, 128 * Lb, 1.0f / ((float)rows * 64.0f));
  }

  // ---- upsample_t: convT 64->64 k4 s2 p1 -> quant-concat ch0-63 ----
  conv_launch(stream, quant_t, F(39), F(40), nullptr, r2, B, 64, Lt, 64, Lb, 4, 2, 1, 64 * Lt, 128 * Lb, 0, 1);

  // ---- dec: convT 128->128 k4 s4 p1; 2 res blocks; wo conv 128->1 k3 p1 ----
  conv_launch(stream, r2, F(41), F(42), nullptr, d2, B, 128, Lb, 128, Ld, 4, 4, 1, 128 * Lb, 128 * Ld, 0, 1);
  res_block(stream, d2, 128 * Ld, F(43), F(44), F(45), F(46), r2, B, Ld);  // r2 free again
  res_block(stream, d2, 128 * Ld, F(47), F(48), F(49), F(50), r2, B, Ld);
  conv_launch(stream, d2, F(51), F(52), nullptr, out, B, 128, Ld, 1, Ld, 3, 1, 1, 128 * Ld, Ld, 0, 0);
#undef F
}